// encoder_31104153157880
// MI455X (gfx1250) — compile-verified
//
#include <hip/hip_runtime.h>

// Problem dims (fixed by reference)
#define VOCAB 32000
#define E_DIM 512
#define H_DIM 1024
#define L_DIM 2
#define S_DIM 512
#define B_DIM 32
#define G4H   (4 * H_DIM)   // 4096 gate width

typedef float v2f __attribute__((ext_vector_type(2)));
typedef float v8f __attribute__((ext_vector_type(8)));

__device__ __forceinline__ float sigmoidf_(float x) {
    return 1.0f / (1.0f + __expf(-x));
}

// ---------------------------------------------------------------------------
// Prep: transpose (rows x cols) -> (cols x rows) so the WMMA B-fragment loads
// (W^T walked along K) are lane-coalesced (16 lanes read 64 contiguous bytes).
// ---------------------------------------------------------------------------
__global__ void transpose_kernel(const float* __restrict__ src,
                                 float* __restrict__ dst,
                                 int rows, int cols) {
    int total = rows * cols;
    for (int idx = blockIdx.x * blockDim.x + threadIdx.x; idx < total;
         idx += gridDim.x * blockDim.x) {
        int r = idx / cols;
        int c = idx - r * cols;
        dst[(size_t)c * rows + r] = src[idx];
    }
}

__global__ void bias_fuse_kernel(const float* __restrict__ a,
                                 const float* __restrict__ b,
                                 float* __restrict__ o, int n) {
    int i = blockIdx.x * blockDim.x + threadIdx.x;
    if (i < n) o[i] = a[i] + b[i];
}

// ---------------------------------------------------------------------------
// One LSTM step for one layer.
//   grid  = H/16 blocks (one 16-wide hidden slice each)
//   block = 256 threads = 8 waves = 2 M-tiles x 4 gates, one 16x16 WMMA tile
//           per wave; full f32 WMMA chain D = A(16x4) * B(4x16) + C.
// Gate fragments meet in LDS so the cell update happens in the same launch.
// ---------------------------------------------------------------------------
__global__ __launch_bounds__(256) void lstm_step_kernel(
    int t, int layer,
    const int*   __restrict__ x,
    const float* __restrict__ embed,
    const float* __restrict__ WtA, int KA,        // input-path weights (KA x 4H)
    const float* __restrict__ WtH,                // recurrent weights  (H  x 4H)
    const float* __restrict__ bias,               // fused b_ih + b_hh  (4H)
    const float* __restrict__ h_init,             // h0 slice for this layer (B x H)
    const float* __restrict__ c_init,             // c0 slice for this layer (B x H)
    float* __restrict__ outs,                     // (S, B, H)
    float* __restrict__ hiddens,                  // (S, L, B, H)
    float* __restrict__ cells)                    // (S, L, B, H)
{
    __shared__ float g_lds[4][B_DIM][16];         // [gate][batch][hidden-in-slice]

    const int lane  = threadIdx.x & 31;
    const int wave  = threadIdx.x >> 5;           // 0..7
    const int mTile = wave & 1;                   // batch rows 0-15 / 16-31
    const int gate  = wave >> 1;                  // 0=i 1=f 2=g 3=o
    const int n0    = blockIdx.x << 4;            // hidden slice base
    const int col   = gate * H_DIM + n0;          // gate column base in 4H
    const int nl    = lane & 15;
    const int kb    = (lane >> 4) << 1;           // lanes 16-31 hold K+2,K+3
    const int mrow  = mTile * 16 + nl;            // batch row this lane loads for A

    // Accumulator init = fused bias, broadcast along M (C/D lanes index N)
    const float bv = bias[col + nl];
    v8f acc;
#pragma unroll
    for (int r = 0; r < 8; ++r) acc[r] = bv;

    // ---- GEMM 1: input path (embedding gather for layer 0, h0(t) for layer 1)
    const float* arow;
    if (layer == 0) {
        arow = embed + (size_t)x[t * B_DIM + mrow] * E_DIM;
    } else {
        arow = hiddens + ((size_t)t * L_DIM + 0) * (B_DIM * H_DIM)
                       + (size_t)mrow * H_DIM;
    }
    const float* bcol = WtA + col + nl;
    for (int k = 0; k < KA; k += 4) {
        v2f a, b;
        a.x = arow[k + kb];
        a.y = arow[k + kb + 1];
        b.x = bcol[(size_t)(k + kb)     * G4H];
        b.y = bcol[(size_t)(k + kb + 1) * G4H];
        acc = __builtin_amdgcn_wmma_f32_16x16x4_f32(
                  false, a, false, b, (short)0, acc, false, false);
    }

    // ---- GEMM 2: recurrent path (h(t-1) of this layer)
    const float* hrow = (t == 0)
        ? (h_init + (size_t)mrow * H_DIM)
        : (hiddens + ((size_t)(t - 1) * L_DIM + layer) * (B_DIM * H_DIM)
                   + (size_t)mrow * H_DIM);
    const float* bcol2 = WtH + col + nl;
    for (int k = 0; k < H_DIM; k += 4) {
        v2f a, b;
        a.x = hrow[k + kb];
        a.y = hrow[k + kb + 1];
        b.x = bcol2[(size_t)(k + kb)     * G4H];
        b.y = bcol2[(size_t)(k + kb + 1) * G4H];
        acc = __builtin_amdgcn_wmma_f32_16x16x4_f32(
                  false, a, false, b, (short)0, acc, false, false);
    }

    // Spill 16x16 fragment to LDS: VGPR r holds M = r + 8*(lane>=16), N = lane&15
#pragma unroll
    for (int r = 0; r < 8; ++r) {
        int m_local = r + ((lane >> 4) << 3);
        g_lds[gate][mTile * 16 + m_local][nl] = acc[r];
    }
    __syncthreads();

    // ---- LSTM cell elementwise update over the 32 x 16 slice
    const size_t stateBase = ((size_t)t * L_DIM + layer) * (B_DIM * H_DIM);
    for (int e = threadIdx.x; e < B_DIM * 16; e += blockDim.x) {
        int m = e >> 4;
        int n = e & 15;
        int hidx = n0 + n;
        float iv = sigmoidf_(g_lds[0][m][n]);
        float fv = sigmoidf_(g_lds[1][m][n]);
        float gv = tanhf    (g_lds[2][m][n]);
        float ov = sigmoidf_(g_lds[3][m][n]);
        float cp = (t == 0)
            ? c_init[(size_t)m * H_DIM + hidx]
            : cells[((size_t)(t - 1) * L_DIM + layer) * (B_DIM * H_DIM)
                    + (size_t)m * H_DIM + hidx];
        float cn = fv * cp + iv * gv;
        float hn = ov * tanhf(cn);
        cells  [stateBase + (size_t)m * H_DIM + hidx] = cn;
        hiddens[stateBase + (size_t)m * H_DIM + hidx] = hn;
        if (layer == L_DIM - 1)
            outs[(size_t)t * (B_DIM * H_DIM) + (size_t)m * H_DIM + hidx] = hn;
    }
}

// ---------------------------------------------------------------------------
extern "C" void kernel_launch(void* const* d_in, const int* in_sizes, int n_in,
                              void* d_out, int out_size, void* d_ws, size_t ws_size,
                              hipStream_t stream) {
    (void)in_sizes; (void)n_in; (void)out_size; (void)ws_size;

    const int*   x     = (const int*)  d_in[0];
    const float* embed = (const float*)d_in[1];
    const float* W_ih0 = (const float*)d_in[2];
    const float* W_hh0 = (const float*)d_in[3];
    const float* b_ih0 = (const float*)d_in[4];
    const float* b_hh0 = (const float*)d_in[5];
    const float* W_ih1 = (const float*)d_in[6];
    const float* W_hh1 = (const float*)d_in[7];
    const float* b_ih1 = (const float*)d_in[8];
    const float* b_hh1 = (const float*)d_in[9];
    const float* h0    = (const float*)d_in[10];  // (L, B, H)
    const float* c0    = (const float*)d_in[11];  // (L, B, H)

    // Workspace: transposed weights (K x 4H) + fused biases, ~59 MB total.
    float* ws     = (float*)d_ws;
    float* Wt_ih0 = ws;                                       // E x 4H
    float* Wt_hh0 = Wt_ih0 + (size_t)E_DIM * G4H;             // H x 4H
    float* Wt_ih1 = Wt_hh0 + (size_t)H_DIM * G4H;             // H x 4H
    float* Wt_hh1 = Wt_ih1 + (size_t)H_DIM * G4H;             // H x 4H
    float* bias0  = Wt_hh1 + (size_t)H_DIM * G4H;             // 4H
    float* bias1  = bias0 + G4H;                              // 4H

    // d_out = outputs (S,1,B,H) ++ hiddens (S,L,B,H) ++ cells (S,L,B,H)
    float* outs    = (float*)d_out;
    float* hiddens = outs + (size_t)S_DIM * B_DIM * H_DIM;
    float* cells   = hiddens + (size_t)S_DIM * L_DIM * B_DIM * H_DIM;

    // Prep: weight transposes + bias fusion (fully parallel, once per call)
    transpose_kernel<<<1024, 256, 0, stream>>>(W_ih0, Wt_ih0, G4H, E_DIM);
    transpose_kernel<<<1024, 256, 0, stream>>>(W_hh0, Wt_hh0, G4H, H_DIM);
    transpose_kernel<<<1024, 256, 0, stream>>>(W_ih1, Wt_ih1, G4H, H_DIM);
    transpose_kernel<<<1024, 256, 0, stream>>>(W_hh1, Wt_hh1, G4H, H_DIM);
    bias_fuse_kernel<<<(G4H + 255) / 256, 256, 0, stream>>>(b_ih0, b_hh0, bias0, G4H);
    bias_fuse_kernel<<<(G4H + 255) / 256, 256, 0, stream>>>(b_ih1, b_hh1, bias1, G4H);

    // Sequential recurrence: one launch per (step, layer). Weights live in the
    // 192 MB L2 across steps; state ping-pongs through the output buffers.
    for (int t = 0; t < S_DIM; ++t) {
        lstm_step_kernel<<<H_DIM / 16, 256, 0, stream>>>(
            t, 0, x, embed, Wt_ih0, E_DIM, Wt_hh0, bias0,
            h0 + 0 * B_DIM * H_DIM, c0 + 0 * B_DIM * H_DIM,
            outs, hiddens, cells);
        lstm_step_kernel<<<H_DIM / 16, 256, 0, stream>>>(
            t, 1, x, embed, Wt_ih1, H_DIM, Wt_hh1, bias1,
            h0 + 1 * B_DIM * H_DIM, c0 + 1 * B_DIM * H_DIM,
            outs, hiddens, cells);
    }
}